// GATConv_49692771614770
// MI455X (gfx1250) — compile-verified
//
#include <hip/hip_runtime.h>
#include <hip/hip_bf16.h>
#include <math.h>

// Problem constants (from reference)
#define NODES   100000
#define EDGES   1600000
#define K_IN    256
#define HEADS   4
#define DHEAD   32
#define HD      128            // HEADS * DHEAD
#define NEG_SLOPE 0.2f

typedef __attribute__((ext_vector_type(2))) float v2f;
typedef __attribute__((ext_vector_type(8))) float v8f;

// -------- device helpers -------------------------------------------------

// Float atomic max via monotonic int reinterpretation:
//  - non-negative floats compare correctly as signed ints (atomicMax)
//  - negative floats compare reversed as unsigned ints (atomicMin)
// Works with init value -inf (0xFF800000).
__device__ __forceinline__ void atomic_max_f32(float* addr, float val) {
  if (!(val < 0.0f)) {
    atomicMax((int*)addr, __float_as_int(val));
  } else {
    atomicMin((unsigned int*)addr, (unsigned int)__float_as_int(val));
  }
}

// Native f32 atomic add (global_atomic_add_f32), relaxed/device scope.
__device__ __forceinline__ void atomic_add_f32(float* addr, float val) {
  __hip_atomic_fetch_add(addr, val, __ATOMIC_RELAXED, __HIP_MEMORY_SCOPE_AGENT);
}

__device__ __forceinline__ float leaky(float x) {
  return x > 0.0f ? x : NEG_SLOPE * x;
}

// reference: emax = where(isfinite(emax), emax, 0). Only -inf can occur.
__device__ __forceinline__ float fix_finite(float x) {
  return x < -1.0e37f ? 0.0f : x;
}

// -------- kernel 0: init out / emax / denom ------------------------------

__global__ void gat_init(float* __restrict__ out,
                         float* __restrict__ emax,
                         float* __restrict__ denom) {
  size_t i = (size_t)blockIdx.x * blockDim.x + threadIdx.x;
  if (i < (size_t)NODES * HD) out[i] = 0.0f;
  if (i < (size_t)NODES * HEADS) {
    emax[i]  = -__builtin_inff();
    denom[i] = 0.0f;
  }
}

// -------- kernel 1: feat_src = feat @ W^T  via V_WMMA_F32_16X16X4_F32 ----
//
// Block = 256 threads = 8 waves. Block b owns node rows [16b, 16b+16).
// Wave w owns output columns [16w, 16w+16)  (HD = 128 = 8 * 16).
// The 16x256 feat tile is staged in LDS once per block (reused by 8 waves).
// Padded LDS row stride (260) so the 16 lanes reading 16 different rows at
// the same k hit 16 different banks (260 % 64 == 4 -> bank = 4*row + k).
//
// ISA 32-bit fragment layouts (wave32):
//  A (16x4, MxK): lane L holds M = L&15;  V0 = K=(L>>4)*2, V1 = K=(L>>4)*2+1
//  B (4x16, KxN): lane L holds N = L&15;  V0 = K=(L>>4)*2, V1 = K=(L>>4)*2+1
//  C/D (16x16):   VGPR r: lanes 0-15 -> M=r, lanes 16-31 -> M=r+8, N = L&15

#define LDS_STRIDE 260

__global__ __launch_bounds__(256) void gat_gemm(const float* __restrict__ feat,
                                                const float* __restrict__ W,
                                                float* __restrict__ feat_src) {
  __shared__ float ldsA[16 * LDS_STRIDE];

  const int m0  = blockIdx.x * 16;
  const int tid = threadIdx.x;

  // Cooperative load: 16 rows x 256 cols = 1024 float4, 4 per thread.
  {
    const float4* gsrc = (const float4*)(feat + (size_t)m0 * K_IN);
#pragma unroll
    for (int i = 0; i < 4; ++i) {
      int q    = tid + i * 256;     // float4 index within tile
      int row  = q >> 6;            // 64 float4 per row
      int col4 = q & 63;
      float4 v = gsrc[row * 64 + col4];
      float* dstp = &ldsA[row * LDS_STRIDE + col4 * 4];
      dstp[0] = v.x; dstp[1] = v.y; dstp[2] = v.z; dstp[3] = v.w;
    }
  }
  __syncthreads();

  const int lane  = tid & 31;
  const int wave  = tid >> 5;
  const int n0    = wave * 16;          // output column tile
  const int mrow  = lane & 15;          // M (for A) / N (for B) index
  const int khalf = (lane >> 4) * 2;    // K sub-offset for this lane half

  const float* arow = &ldsA[mrow * LDS_STRIDE];
  const float* wrow = W + (size_t)(n0 + mrow) * K_IN;   // W[n][k], row-major

  v8f acc = {};
#pragma unroll 4
  for (int k0 = 0; k0 < K_IN; k0 += 4) {
    const int kb = k0 + khalf;
    v2f a = *(const v2f*)(arow + kb);   // 8B aligned (kb even, stride even)
    v2f b = *(const v2f*)(wrow + kb);
    acc = __builtin_amdgcn_wmma_f32_16x16x4_f32(
        /*neg_a=*/false, a, /*neg_b=*/false, b,
        /*c_mod=*/(short)0, acc, /*reuse_a=*/false, /*reuse_b=*/false);
  }

  float* outp = feat_src + (size_t)m0 * HD + n0;
#pragma unroll
  for (int r = 0; r < 8; ++r) {
    const int mm = r + (lane >> 4) * 8;
    outp[(size_t)mm * HD + mrow] = acc[r];
  }
}

// -------- kernel 2: el / er (per node, per head dot with attn vectors) ---
// One wave per (node, head): lane d reads feat_src[node, head*32 + d],
// coalesced 128B, then shuffle-reduce.

__global__ __launch_bounds__(256) void gat_elr(const float* __restrict__ feat_src,
                                               const float* __restrict__ attn_l,
                                               const float* __restrict__ attn_r,
                                               float* __restrict__ el,
                                               float* __restrict__ er) {
  const int tid  = threadIdx.x;
  const int lane = tid & 31;
  const int wave = tid >> 5;
  const int node = blockIdx.x * 2 + (wave >> 2);
  const int head = wave & 3;
  if (node >= NODES) return;

  const float v  = feat_src[(size_t)node * HD + head * DHEAD + lane];
  float pl = v * attn_l[head * DHEAD + lane];
  float pr = v * attn_r[head * DHEAD + lane];
#pragma unroll
  for (int off = 16; off > 0; off >>= 1) {
    pl += __shfl_down(pl, off, 32);
    pr += __shfl_down(pr, off, 32);
  }
  if (lane == 0) {
    el[node * HEADS + head] = pl;
    er[node * HEADS + head] = pr;
  }
}

// -------- kernel 3: segmented max of edge scores over dst ----------------

__global__ __launch_bounds__(256) void gat_edge_max(const int* __restrict__ src,
                                                    const int* __restrict__ dst,
                                                    const float* __restrict__ el,
                                                    const float* __restrict__ er,
                                                    float* __restrict__ emax) {
  const int e = blockIdx.x * blockDim.x + threadIdx.x;
  if (e >= EDGES) return;
  const int s = src[e];
  const int d = dst[e];
  const float4 l = *(const float4*)(el + (size_t)s * HEADS);
  const float4 r = *(const float4*)(er + (size_t)d * HEADS);
  atomic_max_f32(&emax[d * HEADS + 0], leaky(l.x + r.x));
  atomic_max_f32(&emax[d * HEADS + 1], leaky(l.y + r.y));
  atomic_max_f32(&emax[d * HEADS + 2], leaky(l.z + r.z));
  atomic_max_f32(&emax[d * HEADS + 3], leaky(l.w + r.w));
}

// -------- kernel 4: softmax denominator ----------------------------------

__global__ __launch_bounds__(256) void gat_edge_den(const int* __restrict__ src,
                                                    const int* __restrict__ dst,
                                                    const float* __restrict__ el,
                                                    const float* __restrict__ er,
                                                    const float* __restrict__ emax,
                                                    float* __restrict__ denom) {
  const int e = blockIdx.x * blockDim.x + threadIdx.x;
  if (e >= EDGES) return;
  const int s = src[e];
  const int d = dst[e];
  const float4 l = *(const float4*)(el + (size_t)s * HEADS);
  const float4 r = *(const float4*)(er + (size_t)d * HEADS);
  const float4 m = *(const float4*)(emax + (size_t)d * HEADS);
  atomic_add_f32(&denom[d * HEADS + 0], __expf(leaky(l.x + r.x) - fix_finite(m.x)));
  atomic_add_f32(&denom[d * HEADS + 1], __expf(leaky(l.y + r.y) - fix_finite(m.y)));
  atomic_add_f32(&denom[d * HEADS + 2], __expf(leaky(l.z + r.z) - fix_finite(m.z)));
  atomic_add_f32(&denom[d * HEADS + 3], __expf(leaky(l.w + r.w) - fix_finite(m.w)));
}

// -------- kernel 5: attention-weighted aggregation -----------------------
// One wave per edge. Lane l handles columns {l, l+32, l+64, l+96}:
// column c = j*32 + l belongs to head j, so a[j] applies; both the gather
// from feat_src[src] and the atomic scatter to out[dst] are coalesced
// 128B bursts per head.

__global__ __launch_bounds__(256) void gat_edge_agg(const int* __restrict__ src,
                                                    const int* __restrict__ dst,
                                                    const float* __restrict__ el,
                                                    const float* __restrict__ er,
                                                    const float* __restrict__ emax,
                                                    const float* __restrict__ denom,
                                                    const float* __restrict__ feat_src,
                                                    float* __restrict__ out) {
  const int lane = threadIdx.x & 31;
  const long long t = (long long)blockIdx.x * blockDim.x + threadIdx.x;
  const long long e = t >> 5;
  if (e >= EDGES) return;

  const int s = src[e];
  const int d = dst[e];
  const float4 l  = *(const float4*)(el    + (size_t)s * HEADS);
  const float4 r  = *(const float4*)(er    + (size_t)d * HEADS);
  const float4 m  = *(const float4*)(emax  + (size_t)d * HEADS);
  const float4 dn = *(const float4*)(denom + (size_t)d * HEADS);

  float a[4];
  a[0] = __expf(leaky(l.x + r.x) - fix_finite(m.x)) / dn.x;
  a[1] = __expf(leaky(l.y + r.y) - fix_finite(m.y)) / dn.y;
  a[2] = __expf(leaky(l.z + r.z) - fix_finite(m.z)) / dn.z;
  a[3] = __expf(leaky(l.w + r.w) - fix_finite(m.w)) / dn.w;

  const float* fs = feat_src + (size_t)s * HD;
  float*       o  = out      + (size_t)d * HD;
#pragma unroll
  for (int j = 0; j < 4; ++j) {
    const int c = j * DHEAD + lane;
    atomic_add_f32(&o[c], fs[c] * a[j]);
  }
}

// -------- host-side launch ------------------------------------------------

extern "C" void kernel_launch(void* const* d_in, const int* in_sizes, int n_in,
                              void* d_out, int out_size, void* d_ws, size_t ws_size,
                              hipStream_t stream) {
  const float* feat   = (const float*)d_in[0];
  const int*   src    = (const int*)d_in[1];
  const int*   dst    = (const int*)d_in[2];
  const float* W      = (const float*)d_in[3];
  const float* attn_l = (const float*)d_in[4];
  const float* attn_r = (const float*)d_in[5];
  float* out = (float*)d_out;

  // Workspace layout (floats):
  //   feat_src : NODES*HD      (51.2 MB)
  //   el       : NODES*HEADS
  //   er       : NODES*HEADS
  //   emax     : NODES*HEADS
  //   denom    : NODES*HEADS
  float* feat_src = (float*)d_ws;
  float* el    = feat_src + (size_t)NODES * HD;
  float* er    = el    + (size_t)NODES * HEADS;
  float* emax  = er    + (size_t)NODES * HEADS;
  float* denom = emax  + (size_t)NODES * HEADS;

  const int T = 256;

  // 0) init output / softmax state (covers NODES*HD = 12.8M elements)
  gat_init<<<((size_t)NODES * HD + T - 1) / T, T, 0, stream>>>(out, emax, denom);

  // 1) projection: feat_src = feat @ W^T   (WMMA f32 16x16x4)
  gat_gemm<<<NODES / 16, T, 0, stream>>>(feat, W, feat_src);

  // 2) per-node attention logits
  gat_elr<<<(NODES + 1) / 2, T, 0, stream>>>(feat_src, attn_l, attn_r, el, er);

  // 3) segmented softmax over edges, grouped by dst
  gat_edge_max<<<(EDGES + T - 1) / T, T, 0, stream>>>(src, dst, el, er, emax);
  gat_edge_den<<<(EDGES + T - 1) / T, T, 0, stream>>>(src, dst, el, er, emax, denom);

  // 4) weighted aggregation (wave per edge)
  gat_edge_agg<<<((size_t)EDGES * 32 + T - 1) / T, T, 0, stream>>>(
      src, dst, el, er, emax, denom, feat_src, out);
}